// DecodeOnlyMVP_59158879535500
// MI455X (gfx1250) — compile-verified
//
#include <hip/hip_runtime.h>
#include <hip/hip_bf16.h>
#include <math.h>

#define B_    16
#define HID_  4096
#define H_    32
#define D_    128
#define BS_   16
#define MB_   64
#define SCALE_ 0.08838834764831845f   // 1/sqrt(128)
#define EPS_   1e-5f
#define ROPE_BASE_ 10000.0f

// TDM-staged GEMM tiling
#define KT_        32                 // K rows per TDM chunk
#define GCOLS_     64                 // columns per K-split group (4 tiles)
#define LDS_ROW_   80                 // 64 data dwords + 16 pad dwords (TDM pad)

typedef __attribute__((ext_vector_type(2))) float v2f;
typedef __attribute__((ext_vector_type(4))) float v4f;
typedef __attribute__((ext_vector_type(8))) float v8f;
typedef __attribute__((ext_vector_type(4))) unsigned int u32x4;
typedef __attribute__((ext_vector_type(8))) int i32x8;
typedef __attribute__((ext_vector_type(4))) int i32x4;

// ---------------------------------------------------------------------------
// Kernel 1: LayerNorm, one block per row of x (B rows of HID)
// ---------------------------------------------------------------------------
__global__ void ln_kernel(const float* __restrict__ x, float* __restrict__ xn) {
    const int b = blockIdx.x;
    const float* row = x + (size_t)b * HID_;
    float s = 0.f, s2 = 0.f;
    for (int i = threadIdx.x; i < HID_; i += blockDim.x) {
        float v = row[i];
        s += v; s2 += v * v;
    }
    #pragma unroll
    for (int off = 16; off; off >>= 1) {
        s  += __shfl_xor(s,  off, 32);
        s2 += __shfl_xor(s2, off, 32);
    }
    __shared__ float rs[8], rs2[8];
    const int wave = threadIdx.x >> 5;
    if ((threadIdx.x & 31) == 0) { rs[wave] = s; rs2[wave] = s2; }
    __syncthreads();
    if (threadIdx.x == 0) {
        float a = 0.f, c = 0.f;
        for (int w = 0; w < (int)(blockDim.x >> 5); ++w) { a += rs[w]; c += rs2[w]; }
        rs[0] = a; rs2[0] = c;
    }
    __syncthreads();
    const float mean = rs[0] * (1.0f / HID_);
    const float var  = rs2[0] * (1.0f / HID_) - mean * mean;
    const float inv  = rsqrtf(var + EPS_);
    for (int i = threadIdx.x; i < HID_; i += blockDim.x)
        xn[(size_t)b * HID_ + i] = (row[i] - mean) * inv;
}

// ---------------------------------------------------------------------------
// TDM: DMA a KT_ x GCOLS_ fp32 tile of W (row stride HID_) into LDS at
// lds_byte_off, padding each 64-dword row with 16 dwords (LDS row stride 80)
// so the two half-waves of the WMMA B reads land in disjoint bank groups.
// D# per CDNA5 ISA (groups 0/1; 2D tensor -> groups 2,3=0).
// This toolchain exposes the 6-arg builtin: (g0, g1, g2, g3, g4, cpol).
// ---------------------------------------------------------------------------
__device__ __forceinline__ void tdm_load_tile(const float* gptr, unsigned lds_byte_off) {
    unsigned long long ga = (unsigned long long)(uintptr_t)gptr;
    u32x4 g0;
    g0.x = 1u;                                   // count=1 (valid descriptor)
    g0.y = lds_byte_off;                         // lds_addr [63:32]
    g0.z = (unsigned)ga;                         // global_addr [95:64]
    g0.w = (unsigned)(ga >> 32) | 0x80000000u;   // global_addr hi + type=2 [127:126]
    // w0: data_size=4B (2<<16) | pad_enable (1<<20) | pad_interval=64dw (5<<22)
    //     | pad_amount=16dw (15<<25); workgroup_mask=0 (not in cluster)
    i32x8 g1 = {
        (int)((2u << 16) | (1u << 20) | (5u << 22) | (15u << 25)),
        (int)((HID_ & 0xFFFF) << 16),            // tensor_dim0[15:0] -> bits[63:48]
        (int)((HID_ & 0xFFFF) << 16),            // dim0 hi=0; tensor_dim1[15:0] -> bits[95:80]
        (int)(GCOLS_ << 16),                     // dim1 hi=0; tile_dim0 -> bits[127:112]
        (int)KT_,                                // tile_dim1 ; tile_dim2=0
        (int)HID_,                               // tensor_dim0_stride[31:0]
        0, 0                                     // stride hi / dim1_stride unused (2D)
    };
    i32x4 z4 = {0, 0, 0, 0};
    i32x8 z8 = {0, 0, 0, 0, 0, 0, 0, 0};
    __builtin_amdgcn_tensor_load_to_lds(g0, g1, z4, z4, z8, 0);
}

// ---------------------------------------------------------------------------
// One KT_-deep slab of WMMA work: A from global (L2-hot xn/attn row), B from
// the LDS-staged weight tile. Two accumulators break the C->C chain.
// A (16x4 f32): lanes 0-15 hold K=0,1 ; lanes 16-31 hold K=2,3.
// ---------------------------------------------------------------------------
__device__ __forceinline__ void gemm_chunk(const float* __restrict__ arow,
                                           const float* __restrict__ lbuf,
                                           int kglob, int colloc, int r, int koff,
                                           v8f& accA, v8f& accB) {
    #pragma unroll
    for (int kk = 0; kk < KT_; kk += 8) {
        v2f a0 = *(const v2f*)(arow + kglob + kk + koff);
        v2f a1 = *(const v2f*)(arow + kglob + kk + 4 + koff);
        v2f b0, b1;
        b0.x = lbuf[(kk + koff)     * LDS_ROW_ + colloc + r];
        b0.y = lbuf[(kk + koff + 1) * LDS_ROW_ + colloc + r];
        b1.x = lbuf[(kk + 4 + koff) * LDS_ROW_ + colloc + r];
        b1.y = lbuf[(kk + 5 + koff) * LDS_ROW_ + colloc + r];
        accA = __builtin_amdgcn_wmma_f32_16x16x4_f32(false, a0, false, b0, (short)0, accA, false, false);
        accB = __builtin_amdgcn_wmma_f32_16x16x4_f32(false, a1, false, b1, (short)0, accB, false, false);
    }
}

// ---------------------------------------------------------------------------
// TDM double-buffered GEMM body shared by QKV and O-proj kernels.
// Block = 8 waves = 4 column tiles (64 cols) x 2 K-splits (2048 each).
// Wave (grp*4) is the TDM issuer for its group; completion published by
// s_wait_tensorcnt + block barrier. Returns combined v8f for grp==0 waves.
// ---------------------------------------------------------------------------
__device__ __forceinline__ v8f gemm_tdm_body(const float* __restrict__ A,
                                             const float* __restrict__ W,
                                             int colblk, int lane, int wave,
                                             float* __restrict__ stage,   // [2][2][KT_*LDS_ROW_]
                                             v8f*   __restrict__ sacc) {  // [8][32]
    const int tloc = wave & 3;
    const int grp  = wave >> 2;
    const int r    = lane & 15;
    const int koff = (lane >> 4) << 1;
    const int kbeg = grp * (HID_ / 2);
    const int NC   = (HID_ / 2) / KT_;     // 64 chunks

    const float* wbase = W + (size_t)kbeg * HID_ + colblk;
    float* buf0 = stage + (size_t)grp * 2 * (KT_ * LDS_ROW_);
    float* buf1 = buf0 + (KT_ * LDS_ROW_);
    const bool issuer = (tloc == 0);

    if (issuer) tdm_load_tile(wbase, (unsigned)(uintptr_t)buf0);

    v8f acc0 = {}, acc1 = {};
    const float* arow = A + (size_t)r * HID_;

    for (int c = 0; c < NC; ++c) {
        if (issuer) {
            if (c + 1 < NC) {
                tdm_load_tile(wbase + (size_t)(c + 1) * KT_ * HID_,
                              (unsigned)(uintptr_t)((((c + 1) & 1) ? buf1 : buf0)));
                __builtin_amdgcn_s_wait_tensorcnt(1);   // current chunk done, next in flight
            } else {
                __builtin_amdgcn_s_wait_tensorcnt(0);
            }
        }
        __syncthreads();
        gemm_chunk(arow, (c & 1) ? buf1 : buf0, kbeg + c * KT_,
                   tloc * 16, r, koff, acc0, acc1);
        __syncthreads();
    }

    sacc[wave * 32 + lane] = acc0 + acc1;
    __syncthreads();
    return sacc[wave * 32 + lane] + sacc[(wave + 4) * 32 + lane];  // valid for grp==0
}

// ---------------------------------------------------------------------------
// Kernel 2: QKV projection. 768 tiles / 4 per block -> 192 blocks.
// Each block sits inside one of wq/wk/wv (256 tiles each, 4 | 256).
// ---------------------------------------------------------------------------
__global__ void qkv_wmma_kernel(const float* __restrict__ xn,
                                const float* __restrict__ wq,
                                const float* __restrict__ wk,
                                const float* __restrict__ wv,
                                float* __restrict__ qkv) {
    __shared__ float stage[2 * 2 * (KT_ * LDS_ROW_)];
    __shared__ v8f   sacc[8 * 32];

    const int lane = threadIdx.x & 31;
    const int wave = threadIdx.x >> 5;
    const int tile_base = blockIdx.x * 4;
    const int mat    = tile_base >> 8;                 // 0=q 1=k 2=v (uniform)
    const int colblk = (tile_base & 255) << 4;         // 64-wide column stripe

    const float* __restrict__ w = (mat == 0) ? wq : (mat == 1) ? wk : wv;
    float* __restrict__ out = qkv + (size_t)mat * B_ * HID_;

    v8f tot = gemm_tdm_body(xn, w, colblk, lane, wave, stage, sacc);

    if ((wave >> 2) == 0) {
        const int col0 = colblk + (wave & 3) * 16;
        const int r    = lane & 15;
        const int mb   = (lane >> 4) << 3;
        #pragma unroll
        for (int i = 0; i < 8; ++i)
            out[(size_t)(mb + i) * HID_ + col0 + r] = tot[i];
    }
}

// ---------------------------------------------------------------------------
// Kernel 3: RoPE on q and k (in-place in workspace). Block per batch row.
// ---------------------------------------------------------------------------
__global__ void rope_kernel(float* __restrict__ qkv,
                            const int* __restrict__ positions) {
    const int b = blockIdx.x;
    const float pos = (float)positions[b];
    const int half = D_ / 2;
    for (int idx = threadIdx.x; idx < H_ * half; idx += blockDim.x) {
        const int h = idx / half;
        const int i = idx - h * half;
        const float inv_freq = __expf(-__logf(ROPE_BASE_) * (2.0f * i / D_));
        const float ang = pos * inv_freq;
        float c, s;
        __sincosf(ang, &s, &c);
        float* q = qkv + (size_t)b * HID_ + h * D_;
        float* k = qkv + (size_t)B_ * HID_ + (size_t)b * HID_ + h * D_;
        const float q1 = q[i], q2 = q[i + half];
        q[i]        = q1 * c - q2 * s;
        q[i + half] = q2 * c + q1 * s;
        const float k1 = k[i], k2 = k[i + half];
        k[i]        = k1 * c - k2 * s;
        k[i + half] = k2 * c + k1 * s;
    }
}

// ---------------------------------------------------------------------------
// Kernel 4: flash-decode attention. One block per (b,h), 8 waves split the
// context; lane holds float4 of D. K/V cache is streamed once (>L2) -> NT
// b128 loads. Fresh k/v used when t == pos (equivalent to the reference's
// cache scatter, without mutating the input cache).
// ---------------------------------------------------------------------------
__global__ void attn_kernel(const float* __restrict__ qkv,
                            const float* __restrict__ key_cache,
                            const float* __restrict__ value_cache,
                            const int* __restrict__ positions,
                            const int* __restrict__ block_tables,
                            float* __restrict__ attn_out) {
    const int bh   = blockIdx.x;
    const int b    = bh / H_;
    const int h    = bh - b * H_;
    const int lane = threadIdx.x & 31;
    const int wave = threadIdx.x >> 5;
    const int nw   = blockDim.x >> 5;

    const int pos = positions[b];
    const int ctx = pos + 1;

    const float* qrow = qkv + (size_t)b * HID_ + h * D_;
    const float* knew = qkv + (size_t)B_ * HID_ + (size_t)b * HID_ + h * D_;
    const float* vnew = qkv + (size_t)2 * B_ * HID_ + (size_t)b * HID_ + h * D_;

    const v4f q4 = *(const v4f*)(qrow + lane * 4);

    float m = -__builtin_inff();
    float l = 0.f;
    v4f acc = {};

    for (int t = wave; t < ctx; t += nw) {
        v4f k4, v4;
        if (t == pos) {
            k4 = *(const v4f*)(knew + lane * 4);
            v4 = *(const v4f*)(vnew + lane * 4);
        } else {
            const int blk = block_tables[b * MB_ + (t >> 4)];
            const size_t base = (((size_t)blk * H_ + h) * BS_ + (t & 15)) * D_;
            k4 = __builtin_nontemporal_load((const v4f*)(key_cache + base + lane * 4));
            v4 = __builtin_nontemporal_load((const v4f*)(value_cache + base + lane * 4));
        }
        float s = q4.x * k4.x + q4.y * k4.y + q4.z * k4.z + q4.w * k4.w;
        #pragma unroll
        for (int off = 16; off; off >>= 1) s += __shfl_xor(s, off, 32);
        s *= SCALE_;

        const float mnew = fmaxf(m, s);
        const float sc   = __expf(m - mnew);
        const float p    = __expf(s - mnew);
        acc = acc * sc + p * v4;
        l = l * sc + p;
        m = mnew;
    }

    __shared__ float sm[8], sl[8];
    __shared__ v4f sacc[8][32];
    sm[wave] = m;
    sl[wave] = l;
    sacc[wave][lane] = acc;
    __syncthreads();

    if (wave == 0) {
        float M = -__builtin_inff();
        for (int w = 0; w < nw; ++w) M = fmaxf(M, sm[w]);
        float L = 0.f;
        v4f o = {};
        for (int w = 0; w < nw; ++w) {
            const float f = __expf(sm[w] - M);
            L += f * sl[w];
            o = o + f * sacc[w][lane];
        }
        const float invL = 1.0f / L;
        *(v4f*)(attn_out + (size_t)bh * D_ + lane * 4) = o * invL;
    }
}

// ---------------------------------------------------------------------------
// Kernel 5: output projection + residual. 256 tiles / 4 -> 64 blocks.
// ---------------------------------------------------------------------------
__global__ void oproj_wmma_kernel(const float* __restrict__ attn,
                                  const float* __restrict__ wo,
                                  const float* __restrict__ x,
                                  float* __restrict__ out) {
    __shared__ float stage[2 * 2 * (KT_ * LDS_ROW_)];
    __shared__ v8f   sacc[8 * 32];

    const int lane = threadIdx.x & 31;
    const int wave = threadIdx.x >> 5;
    const int colblk = blockIdx.x * 64;

    v8f tot = gemm_tdm_body(attn, wo, colblk, lane, wave, stage, sacc);

    if ((wave >> 2) == 0) {
        const int col0 = colblk + (wave & 3) * 16;
        const int r    = lane & 15;
        const int mb   = (lane >> 4) << 3;
        #pragma unroll
        for (int i = 0; i < 8; ++i) {
            const int mrow = mb + i;
            out[(size_t)mrow * HID_ + col0 + r] =
                x[(size_t)mrow * HID_ + col0 + r] + tot[i];
        }
    }
}

// ---------------------------------------------------------------------------
// Host launcher
// ---------------------------------------------------------------------------
extern "C" void kernel_launch(void* const* d_in, const int* in_sizes, int n_in,
                              void* d_out, int out_size, void* d_ws, size_t ws_size,
                              hipStream_t stream) {
    const float* x            = (const float*)d_in[0];
    const int*   positions    = (const int*)d_in[1];
    const float* key_cache    = (const float*)d_in[2];
    const float* value_cache  = (const float*)d_in[3];
    const int*   block_tables = (const int*)d_in[4];
    const float* wq           = (const float*)d_in[5];
    const float* wk           = (const float*)d_in[6];
    const float* wv           = (const float*)d_in[7];
    const float* wo           = (const float*)d_in[8];
    float* out = (float*)d_out;

    // Workspace layout (floats): xn[16*4096] | qkv[3*16*4096] | attn[16*4096]
    float* ws   = (float*)d_ws;
    float* xn   = ws;
    float* qkv  = ws + (size_t)B_ * HID_;
    float* attn = ws + (size_t)4 * B_ * HID_;

    ln_kernel<<<B_, 256, 0, stream>>>(x, xn);
    qkv_wmma_kernel<<<192, 256, 0, stream>>>(xn, wq, wk, wv, qkv);    // 768 tiles, TDM-staged
    rope_kernel<<<B_, 256, 0, stream>>>(qkv, positions);
    attn_kernel<<<B_ * H_, 256, 0, stream>>>(qkv, key_cache, value_cache,
                                             positions, block_tables, attn);
    oproj_wmma_kernel<<<64, 256, 0, stream>>>(attn, wo, x, out);      // 256 tiles, TDM-staged
}